// GraphAttentionLayer_6064493821983
// MI455X (gfx1250) — compile-verified
//
#include <hip/hip_runtime.h>
#include <math.h>

// GAT layer for MI455X (gfx1250, wave32).
// B=2, N=1024, IN=256, H=64.
// Pipeline: [WMMA proj x2] -> [e-score VALU tile kernel] -> [column softmax]
//           -> [WMMA attention GEMM with folded 1/colsum scaling + ELU epilogue]

#define GN   1024
#define GIN  256
#define GH   64
#define GB   2
#define GALPHA 0.2f
#define GNEGINF (-9e15f)

typedef float v2f __attribute__((ext_vector_type(2)));
typedef float v8f __attribute__((ext_vector_type(8)));

// ---------------------------------------------------------------------------
// Generic WMMA f32 GEMM:  C[M x 64] = A[M x K] @ B[K x 64]  (row-major f32)
// 128 threads = 4 waves; each block does a 16-row x 64-col output block,
// wave w owns output columns [16w, 16w+16). K staged in 64-deep slices.
// Optional per-K-row scale on B (folds softmax 1/colsum into the B stage),
// optional ELU epilogue.
// ---------------------------------------------------------------------------
__global__ __launch_bounds__(128) void wmma_gemm_n64(
    const float* __restrict__ A, int lda, long long sA,
    const float* __restrict__ Bm, long long sB,
    const float* __restrict__ bscale, long long sScale,
    float* __restrict__ C, long long sC,
    int K, int epi)
{
  __shared__ float As[16][68];   // stride 68: b64 frag reads hit all 64 banks once
  __shared__ float Bs[64][72];   // stride 72: half-wave rows 16 banks apart

  const int bt = blockIdx.y;
  A  += (size_t)bt * sA;
  Bm += (size_t)bt * sB;
  C  += (size_t)bt * sC;
  const bool hasScale = (bscale != nullptr);
  if (hasScale) bscale += (size_t)bt * sScale;

  const int row0 = blockIdx.x * 16;
  const int t    = threadIdx.x;
  const int lane = t & 31;
  const int wv   = t >> 5;          // wave -> 16-col slice
  const int half = lane >> 4;
  const int l16  = lane & 15;
  const int ncol = wv * 16 + l16;

  v8f acc = {};

  for (int kb = 0; kb < K; kb += 64) {
    // ---- stage A tile: 16 x 64 floats = 256 float4, 2 per thread
#pragma unroll
    for (int r = 0; r < 2; ++r) {
      int idx  = t + r * 128;
      int arow = idx >> 4, ac4 = idx & 15;
      float4 v = *(const float4*)(A + (size_t)(row0 + arow) * lda + kb + ac4 * 4);
      *(float4*)&As[arow][ac4 * 4] = v;
    }
    // ---- stage B tile: 64 x 64 floats = 1024 float4, 8 per thread
#pragma unroll
    for (int r = 0; r < 8; ++r) {
      int idx  = t + r * 128;
      int brow = idx >> 4, bc4 = idx & 15;
      float4 v = *(const float4*)(Bm + (size_t)(kb + brow) * GH + bc4 * 4);
      if (hasScale) {
        float sc = bscale[kb + brow];
        v.x *= sc; v.y *= sc; v.z *= sc; v.w *= sc;
      }
      *(float4*)&Bs[brow][bc4 * 4] = v;
    }
    __syncthreads();

    // ---- 16 chained V_WMMA_F32_16X16X4_F32 over this 64-deep slice
#pragma unroll
    for (int kk = 0; kk < 64; kk += 4) {
      v2f af, bf;
      // A 16x4 frag: lane l16 -> row, half selects K pair (contiguous -> b64)
      af.x = As[l16][kk + 2 * half];
      af.y = As[l16][kk + 2 * half + 1];
      // B 4x16 frag: VGPR0 = rows K=kk / kk+2, VGPR1 = rows kk+1 / kk+3
      bf.x = Bs[kk + 2 * half][ncol];
      bf.y = Bs[kk + 2 * half + 1][ncol];
      acc = __builtin_amdgcn_wmma_f32_16x16x4_f32(
          /*neg_a=*/false, af, /*neg_b=*/false, bf,
          /*c_mod=*/(short)0, acc, /*reuse_a=*/false, /*reuse_b=*/false);
    }
    __syncthreads();
  }

  // ---- epilogue + store: VGPR r -> output row r (lanes 0-15) / r+8 (16-31)
#pragma unroll
  for (int r = 0; r < 8; ++r) {
    int orow = r + 8 * half;
    float v = acc[r];
    if (epi) v = (v > 0.0f) ? v : (__expf(v) - 1.0f);   // ELU (alpha=1)
    C[(size_t)(row0 + orow) * GH + ncol] = v;
  }
}

// ---------------------------------------------------------------------------
// e[b,i,j] = adj>0 ? sum_h a[h]*lrelu(Wh2[b,i,h]+Wh1[b,j,h]) : -9e15
// 256 threads tile 32 j x 32 i; each thread: 1 j, 4 i's, h-loop in float4.
// ---------------------------------------------------------------------------
__global__ __launch_bounds__(256) void gat_escore(
    const float* __restrict__ Wh1, const float* __restrict__ Wh2,
    const float* __restrict__ av,  const float* __restrict__ adj,
    float* __restrict__ E)
{
  __shared__ float s1[32][68];   // Wh1 rows for j-tile
  __shared__ float s2[32][68];   // Wh2 rows for i-tile
  __shared__ float sa[64];

  const int b  = blockIdx.z;
  const int j0 = blockIdx.x * 32;
  const int i0 = blockIdx.y * 32;
  const int t  = threadIdx.x;
  const int tx = t & 31;   // j within tile (lane)
  const int ty = t >> 5;   // i group (0..7), 4 rows each

  {
    const float4* w1p = (const float4*)(Wh1 + (size_t)b * GN * GH + (size_t)j0 * GH);
    const float4* w2p = (const float4*)(Wh2 + (size_t)b * GN * GH + (size_t)i0 * GH);
#pragma unroll
    for (int r = 0; r < 2; ++r) {
      int idx = t + r * 256;          // 512 float4 per matrix
      int row = idx >> 4, c4 = idx & 15;
      *(float4*)&s1[row][c4 * 4] = w1p[row * 16 + c4];
      *(float4*)&s2[row][c4 * 4] = w2p[row * 16 + c4];
    }
    if (t < 16) *(float4*)&sa[t * 4] = ((const float4*)av)[t];
  }
  __syncthreads();

  float acc[4] = {0.f, 0.f, 0.f, 0.f};
#pragma unroll
  for (int hb = 0; hb < 16; ++hb) {
    float4 w1 = *(const float4*)&s1[tx][hb * 4];
    float4 aa = *(const float4*)&sa[hb * 4];
#pragma unroll
    for (int s = 0; s < 4; ++s) {
      float4 w2 = *(const float4*)&s2[ty * 4 + s][hb * 4];
      float x;
      x = w1.x + w2.x; acc[s] += aa.x * fmaxf(x, GALPHA * x);
      x = w1.y + w2.y; acc[s] += aa.y * fmaxf(x, GALPHA * x);
      x = w1.z + w2.z; acc[s] += aa.z * fmaxf(x, GALPHA * x);
      x = w1.w + w2.w; acc[s] += aa.w * fmaxf(x, GALPHA * x);
    }
  }

  const int j = j0 + tx;
#pragma unroll
  for (int s = 0; s < 4; ++s) {
    int i = i0 + ty * 4 + s;
    size_t idx = (size_t)b * GN * GN + (size_t)i * GN + j;
    E[idx] = (adj[idx] > 0.0f) ? acc[s] : GNEGINF;
  }
}

// ---------------------------------------------------------------------------
// Softmax over axis=1 (over i, per (b,j) column). In-place E -> p = exp(e-max),
// writes colinv[b*N+j] = 1/sum_i p. 8 threads per column, 32 columns per block.
// ---------------------------------------------------------------------------
__global__ __launch_bounds__(256) void gat_softmax_col(
    float* __restrict__ E, float* __restrict__ colinv)
{
  __shared__ float red[8][33];
  const int t = threadIdx.x;
  const int c = t & 31;          // column slot (consecutive j -> coalesced)
  const int s = t >> 5;          // i-slice 0..7 (128 rows each)
  const int bj = blockIdx.x * 32 + c;
  const int b  = bj >> 10;
  const int j  = bj & (GN - 1);
  float* col = E + (size_t)b * GN * GN + j;

  float m = -3.4e38f;
  for (int i = s * 128; i < s * 128 + 128; ++i)
    m = fmaxf(m, col[(size_t)i * GN]);
  red[s][c] = m;
  __syncthreads();
  if (s == 0) {
    float mm = red[0][c];
#pragma unroll
    for (int k = 1; k < 8; ++k) mm = fmaxf(mm, red[k][c]);
    red[0][c] = mm;
  }
  __syncthreads();
  const float mm = red[0][c];

  float sum = 0.0f;
  for (int i = s * 128; i < s * 128 + 128; ++i) {
    float p = __expf(col[(size_t)i * GN] - mm);
    col[(size_t)i * GN] = p;
    sum += p;
  }
  __syncthreads();               // protect red[] reuse
  red[s][c] = sum;
  __syncthreads();
  if (s == 0) {
    float ss = 0.0f;
#pragma unroll
    for (int k = 0; k < 8; ++k) ss += red[k][c];
    colinv[bj] = 1.0f / ss;
  }
}

// ---------------------------------------------------------------------------
extern "C" void kernel_launch(void* const* d_in, const int* in_sizes, int n_in,
                              void* d_out, int out_size, void* d_ws, size_t ws_size,
                              hipStream_t stream) {
  const float* h   = (const float*)d_in[0];   // [2,1024,256]
  const float* adj = (const float*)d_in[1];   // [2,1024,1024]
  const float* W1  = (const float*)d_in[2];   // [256,64]
  const float* W2  = (const float*)d_in[3];   // [256,64]
  const float* av  = (const float*)d_in[4];   // [64]
  float* out = (float*)d_out;                 // [2,1024,64]

  float* ws   = (float*)d_ws;
  float* Wh1  = ws;                    // 131072 f
  float* Wh2  = ws + 131072;           // 131072 f
  float* P    = ws + 262144;           // 2097152 f (e, then p, in place)
  float* cinv = ws + 2359296;          // 2048 f

  // Projections: [2048 x 256] @ [256 x 64] (h flattened over B,N)
  wmma_gemm_n64<<<dim3(128, 1), 128, 0, stream>>>(
      h, GIN, 0, W1, 0, nullptr, 0, Wh1, 0, GIN, /*epi=*/0);
  wmma_gemm_n64<<<dim3(128, 1), 128, 0, stream>>>(
      h, GIN, 0, W2, 0, nullptr, 0, Wh2, 0, GIN, /*epi=*/0);

  // Masked attention logits e[b,i,j]
  gat_escore<<<dim3(GN / 32, GN / 32, GB), 256, 0, stream>>>(Wh1, Wh2, av, adj, P);

  // Column softmax (axis=1), in-place exp + 1/colsum
  gat_softmax_col<<<(GB * GN) / 32, 256, 0, stream>>>(P, cinv);

  // h' = p @ (Wh1 * colinv[j])  then ELU.  Per batch: [1024x1024]@[1024x64]
  wmma_gemm_n64<<<dim3(GN / 16, GB), 128, 0, stream>>>(
      P, GN, (long long)GN * GN, Wh1, (long long)GN * GH,
      cinv, (long long)GN, out, (long long)GN * GH, GN, /*epi=*/1);
}